// LiquidBrain_9448928051674
// MI455X (gfx1250) — compile-verified
//
#include <hip/hip_runtime.h>
#include <hip/hip_bf16.h>

typedef float v2f __attribute__((ext_vector_type(2)));
typedef float v8f __attribute__((ext_vector_type(8)));

__device__ __forceinline__ float sigf(float x) { return 1.f / (1.f + expf(-x)); }

__device__ __forceinline__ float apply_act(float x, int act) {
  if (act == 1) return sigf(x);                               // sigmoid
  if (act == 2) return 0.5f * x * (1.f + erff(x * 0.70710678118654752f)); // exact gelu
  return x;
}

// ---------------------------------------------------------------------------
// Tiled WMMA fp32 GEMM: C = act(A[MxK] * W[KxN] + bias), optional group strides
// block = 256 threads = 8 wave32, block tile 32x64, K-step 32 via LDS.
// Each wave computes one 16x16 tile with V_WMMA_F32_16X16X4_F32.
// Staging: float4 (b128) global loads + b128 LDS stores, software-pipelined
// one K-tile ahead; weight stream prefetched 8 tiles ahead into L2.
// M % 32 == 0, N % 64 == 0, K % 32 == 0 (true for every call below).
// ---------------------------------------------------------------------------
__global__ __launch_bounds__(256) void gemm_wmma(
    const float* __restrict__ A, const float* __restrict__ W,
    const float* __restrict__ bias, float* __restrict__ C,
    int M, int N, int K, int lda, int ldw, int ldc,
    long long aGroupStride, long long cGroupStride, int act) {
  __shared__ __align__(16) float As[32][36];  // row stride 36: 16B-aligned rows,
                                              // 36*r mod 64 distinct -> conflict-free
  __shared__ __align__(16) float Ws[32][64];
  const int g = blockIdx.z;
  A += (size_t)g * (size_t)aGroupStride;
  C += (size_t)g * (size_t)cGroupStride;
  const int m0 = blockIdx.y * 32;
  const int n0 = blockIdx.x * 64;
  const int tid = threadIdx.x;
  const int wave = tid >> 5, lane = tid & 31;
  const int wm = wave & 1, wn = wave >> 1;     // 2 x 4 wave tile grid
  const int half = lane >> 4, l15 = lane & 15;

  // per-thread staging coordinates (float4 granularity)
  const int ar = tid >> 3, ac = (tid & 7) * 4;        // A tile: 32 rows x 8 quads
  const int wr = tid >> 4, wc = (tid & 15) * 4;       // W tile: 16 rows x 16 quads, x2
  const float* ap  = A + (size_t)(m0 + ar) * lda + ac;
  const float* wp0 = W + (size_t)wr * ldw + (n0 + wc);
  const float* wp1 = W + (size_t)(wr + 16) * ldw + (n0 + wc);
  const float* wpf = wp0 + (size_t)256 * ldw;         // prefetch 8 K-tiles ahead
  const size_t wstep = (size_t)32 * ldw;

  v8f acc = {0.f, 0.f, 0.f, 0.f, 0.f, 0.f, 0.f, 0.f};

  // preload first tile into registers
  float4 av  = *(const float4*)ap;
  float4 w0v = *(const float4*)wp0;
  float4 w1v = *(const float4*)wp1;
  ap += 32; wp0 += wstep; wp1 += wstep;

  for (int k0 = 0; k0 < K; k0 += 32) {
    __syncthreads();                 // previous compute done before overwrite
    *(float4*)&As[ar][ac] = av;
    *(float4*)&Ws[wr][wc] = w0v;
    *(float4*)&Ws[wr + 16][wc] = w1v;
    __syncthreads();
    if (k0 + 32 < K) {               // pipeline: fetch next tile during compute
      av  = *(const float4*)ap;
      w0v = *(const float4*)wp0;
      w1v = *(const float4*)wp1;
      ap += 32; wp0 += wstep; wp1 += wstep;
      if (k0 + 256 < K) {
        __builtin_prefetch((const void*)wpf, 0, 1);   // global_prefetch_b8 (L2)
        wpf += wstep;
      }
    }
#pragma unroll
    for (int kk = 0; kk < 32; kk += 4) {
      const int kb = kk + 2 * half;
      v2f a, b;
      a.x = As[wm * 16 + l15][kb];
      a.y = As[wm * 16 + l15][kb + 1];
      b.x = Ws[kb][wn * 16 + l15];
      b.y = Ws[kb + 1][wn * 16 + l15];
      acc = __builtin_amdgcn_wmma_f32_16x16x4_f32(false, a, false, b,
                                                  (short)0, acc, false, false);
    }
  }

  const int col = n0 + wn * 16 + l15;
  const float bv = bias ? bias[col] : 0.f;
  float* cp = C + (size_t)(m0 + wm * 16 + half * 8) * ldc + col;
#pragma unroll
  for (int r = 0; r < 8; ++r) {
    *cp = apply_act(acc[r] + bv, act);
    cp += ldc;
  }
}

// ---------------------------------------------------------------------------
// Small kernels
// ---------------------------------------------------------------------------
__global__ void k_token_router(const int* x_t, const int* role_id,
                               const float* emb, const float* role_emb,
                               const float* router_w,
                               float* token, float* rw, float* active) {
  int b = blockIdx.x, tid = threadIdx.x;       // block 512
  __shared__ float tok[512];
  __shared__ float sc[16];
  int t = x_t[b], r = role_id[0];
  float tv = emb[(size_t)t * 512 + tid] + role_emb[(size_t)r * 512 + tid];
  tok[tid] = tv;
  token[(size_t)b * 512 + tid] = tv;
  __syncthreads();
  if (tid < 16) {
    float acc = 0.f;
    for (int d = 0; d < 512; ++d) acc += tok[d] * router_w[d * 16 + tid];
    sc[tid] = acc;
  }
  __syncthreads();
  if (tid == 0) {
    float mx = sc[0];
    for (int g = 1; g < 16; ++g) mx = fmaxf(mx, sc[g]);
    float s[16], sum = 0.f;
    for (int g = 0; g < 16; ++g) { s[g] = expf(sc[g] - mx); sum += s[g]; }
    for (int g = 0; g < 16; ++g) s[g] /= sum;
    int i1 = 0;
    for (int g = 1; g < 16; ++g) if (s[g] > s[i1]) i1 = g;
    int i2 = (i1 == 0) ? 1 : 0;
    for (int g = 0; g < 16; ++g) if (g != i1 && s[g] > s[i2]) i2 = g;
    float asum = s[i1] + s[i2] + 1e-8f;
    for (int g = 0; g < 16; ++g) {
      float a = (g == i1 || g == i2) ? 1.f : 0.f;
      active[b * 16 + g] = a;
      rw[b * 16 + g] = a * s[g] / asum;
    }
  }
}

// LayerNorm over last dim; optional per-row mask (index = row>>4), block=256
__global__ void ln_k(const float* x, const float* mask,
                     const float* gw, const float* bw,
                     float* out, int D, float eps) {
  int row = blockIdx.x, tid = threadIdx.x;
  float m = mask ? mask[row >> 4] : 1.f;
  float xs[2];
  int cnt = 0;
  float sum = 0.f;
  for (int d = tid; d < D; d += 256) {
    float v = x[(size_t)row * D + d] * m;
    xs[cnt++] = v;
    sum += v;
  }
  __shared__ float red[256];
  red[tid] = sum; __syncthreads();
  for (int s = 128; s > 0; s >>= 1) { if (tid < s) red[tid] += red[tid + s]; __syncthreads(); }
  float mu = red[0] / (float)D;
  __syncthreads();
  float vs = 0.f; cnt = 0;
  for (int d = tid; d < D; d += 256) { float dv = xs[cnt++] - mu; vs += dv * dv; }
  red[tid] = vs; __syncthreads();
  for (int s = 128; s > 0; s >>= 1) { if (tid < s) red[tid] += red[tid + s]; __syncthreads(); }
  float rstd = rsqrtf(red[0] / (float)D + eps);
  cnt = 0;
  for (int d = tid; d < D; d += 256)
    out[(size_t)row * D + d] = (xs[cnt++] - mu) * rstd * gw[d] + bw[d];
}

// x / max(||x||, eps), rows of 64, block = 64
__global__ void rownorm_k(float* x, float eps) {
  int row = blockIdx.x, tid = threadIdx.x;
  float v = x[(size_t)row * 64 + tid];
  __shared__ float red[64];
  red[tid] = v * v; __syncthreads();
  for (int s = 32; s > 0; s >>= 1) { if (tid < s) red[tid] += red[tid + s]; __syncthreads(); }
  float n = sqrtf(red[0]);
  x[(size_t)row * 64 + tid] = v / fmaxf(n, eps);
}

__global__ void k_grp(const float* mnorm, float* grp) {
  int bg = blockIdx.x, b = bg >> 4, g = bg & 15;
  for (int d = threadIdx.x; d < 512; d += 256) {
    float acc = 0.f;
    for (int n = 0; n < 16; ++n)
      acc += mnorm[(size_t)(b * 256 + g * 16 + n) * 512 + d];
    grp[(size_t)bg * 512 + d] = acc * (1.f / 16.f);
  }
}

__global__ void k_glob(const float* grp, const float* rw, float* glob) {
  int b = blockIdx.x;
  for (int d = threadIdx.x; d < 512; d += 256) {
    float acc = 0.f;
    for (int g = 0; g < 16; ++g)
      acc += grp[(size_t)(b * 16 + g) * 512 + d] * rw[b * 16 + g];
    glob[(size_t)b * 512 + d] = acc;
  }
}

// naive C[M,N] = act(A@W + bias); grid.x = rows, block = N
__global__ void naive_mm(const float* A, const float* W, const float* bias,
                         float* C, int K, int N, int act) {
  int row = blockIdx.x, col = threadIdx.x;
  float acc = bias ? bias[col] : 0.f;
  const float* a = A + (size_t)row * K;
  for (int k = 0; k < K; ++k) acc += a[k] * W[(size_t)k * N + col];
  C[(size_t)row * N + col] = apply_act(acc, act);
}

__global__ void k_inter_gate(const float* grp, const float* routed,
                             const float* W, const float* bias, float* inter) {
  int bg = blockIdx.x, b = bg >> 4, d = threadIdx.x; // block 512
  float acc = bias[d];
  const float* gr = grp + (size_t)bg * 512;
  const float* ro = routed + (size_t)b * 512;
  for (int k = 0; k < 512; ++k) acc += gr[k] * W[(size_t)k * 512 + d];
  for (int k = 0; k < 512; ++k) acc += ro[k] * W[(size_t)(512 + k) * 512 + d];
  inter[(size_t)bg * 512 + d] = sigf(acc) * ro[d];
}

// Hebbian fast-weight update, one block per (b,g), block = 256
__global__ __launch_bounds__(256) void k_hebbian(
    const float* mnorm, const float* qb, const float* kb, const float* vb,
    const float* Mprev, const float* decay_w, const float* decay_b,
    const float* active, float* Mnext_out, float* hebb) {
  int bg = blockIdx.x, b = bg >> 4, g = bg & 15, tid = threadIdx.x;
  __shared__ float kk[16][64], vv[16][64], qq[16][64];
  __shared__ float Mp[64][64], Mu[64][64];
  __shared__ float vp[16][64];
  __shared__ float red[256];
  __shared__ float gamma_s;
  int rowbase = b * 256 + g * 16;
#pragma unroll
  for (int i = 0; i < 4; ++i) {
    int e = tid + 256 * i, n = e >> 6, j = e & 63;
    size_t idx = (size_t)(rowbase + n) * 64 + j;
    qq[n][j] = qb[idx]; kk[n][j] = kb[idx]; vv[n][j] = vb[idx];
  }
#pragma unroll
  for (int i = 0; i < 16; ++i) {
    int e = tid + 256 * i;
    Mp[e >> 6][e & 63] = Mprev[(size_t)bg * 4096 + e];
  }
  float dacc = 0.f;
  for (int e = tid; e < 16 * 512; e += 256) {
    int n = e >> 9, d = e & 511;
    dacc += mnorm[(size_t)(rowbase + n) * 512 + d] * decay_w[d];
  }
  red[tid] = dacc; __syncthreads();
  for (int s = 128; s > 0; s >>= 1) { if (tid < s) red[tid] += red[tid + s]; __syncthreads(); }
  if (tid == 0) {
    float dm = red[0] * (1.f / 16.f) + decay_b[0];
    float lin = -3.f + (11.f / 15.f) * (float)g;  // linspace(-3, 8, 16)
    gamma_s = sigf(dm + lin);
  }
  __syncthreads();
#pragma unroll
  for (int s = 0; s < 4; ++s) {                 // v_pred[n][i] = sum_j k[n][j]*Mp[i][j]
    int e = tid + 256 * s, n = e >> 6, i = e & 63;
    float acc = 0.f;
    for (int j = 0; j < 64; ++j) acc += kk[n][j] * Mp[i][j];
    vp[n][i] = acc;
  }
  __syncthreads();
  float gam = gamma_s, am = active[bg];
#pragma unroll
  for (int s = 0; s < 16; ++s) {                // M_upd, M_next
    int e = tid + 256 * s, i = e >> 6, j = e & 63;
    float acc = 0.f;
    for (int n = 0; n < 16; ++n) acc += (vv[n][i] - vp[n][i]) * kk[n][j];
    float mu = gam * Mp[i][j] + acc * (1.f / 16.f);
    mu = fminf(fmaxf(mu, -10.f), 10.f);
    Mu[i][j] = mu;
    Mnext_out[(size_t)bg * 4096 + e] = (am > 0.5f) ? mu : Mp[i][j];
  }
  __syncthreads();
#pragma unroll
  for (int s = 0; s < 4; ++s) {                 // hebb[n][i] = sum_j q[n][j]*Mu[i][j]
    int e = tid + 256 * s, n = e >> 6, i = e & 63;
    float acc = 0.f;
    for (int j = 0; j < 64; ++j) acc += qq[n][j] * Mu[i][j];
    hebb[(size_t)(rowbase + n) * 64 + i] = acc;
  }
}

__global__ void k_concat(const float* token, const float* neuron_emb,
                         const float* inter, const float* intra,
                         const float* hebb_out, const float* mnorm, float* comb) {
  int row = blockIdx.x, b = row >> 8, n = row & 255, g = n >> 4, d = threadIdx.x;
  size_t co = (size_t)row * 2560;
  comb[co + d]        = token[(size_t)b * 512 + d] + neuron_emb[(size_t)n * 512 + d];
  comb[co + 512 + d]  = inter[(size_t)(b * 16 + g) * 512 + d];
  comb[co + 1024 + d] = intra[(size_t)row * 512 + d];
  comb[co + 1536 + d] = hebb_out[(size_t)row * 512 + d];
  comb[co + 2048 + d] = mnorm[(size_t)row * 512 + d];
}

__global__ void k_ops_update(const float* h2, const float* mlpout,
                             const float* fb, const float* ib,
                             const float* mnorm, const float* m_prev,
                             const float* active, const float* op_logits,
                             float* m_next) {
  int row = blockIdx.x, b = row >> 8, n = row & 255, g = n >> 4, d = threadIdx.x;
  float l0 = op_logits[g * 5 + 0], l1 = op_logits[g * 5 + 1], l2 = op_logits[g * 5 + 2],
        l3 = op_logits[g * 5 + 3], l4 = op_logits[g * 5 + 4];
  float mx = fmaxf(fmaxf(fmaxf(l0, l1), fmaxf(l2, l3)), l4);
  float e0 = expf(l0 - mx), e1 = expf(l1 - mx), e2 = expf(l2 - mx),
        e3 = expf(l3 - mx), e4 = expf(l4 - mx);
  float es = e0 + e1 + e2 + e3 + e4;
  size_t ho = (size_t)row * 2560;
  float o1 = h2[ho + d], o2 = h2[ho + 512 + d], o3 = h2[ho + 1024 + d],
        o4 = h2[ho + 1536 + d], o5 = h2[ho + 2048 + d];
  float mn = mnorm[(size_t)row * 512 + d];
  float ops = (e0 / es) * tanhf(o1) + (e1 / es) * (sigf(o2) * mn) +
              (e2 / es) * cosf(o3) + (e3 / es) * (o4 * sigf(o4)) +
              (e4 / es) * (-fmaxf(o5, 0.f));
  size_t idx = (size_t)row * 512 + d;
  float mp = m_prev[idx];
  float mu = fb[idx] * mp + ib[idx] * (ops + mlpout[idx]);
  mu = fminf(fmaxf(mu, -20.f), 20.f);
  m_next[idx] = (active[b * 16 + g] > 0.5f) ? mu : mp;
}

__global__ void k_readout_att(const float* mn, const float* rq, float* att) {
  int b = blockIdx.x, n = threadIdx.x;  // block 256
  float acc = 0.f;
  for (int d = 0; d < 512; ++d) acc += rq[d] * mn[(size_t)(b * 256 + n) * 512 + d];
  acc *= 0.04419417382415922f;  // 1/sqrt(512)
  __shared__ float red[256];
  __shared__ float smax, ssum;
  red[n] = acc; __syncthreads();
  if (n == 0) { float m = red[0]; for (int t = 1; t < 256; ++t) m = fmaxf(m, red[t]); smax = m; }
  __syncthreads();
  float ev = expf(acc - smax);
  red[n] = ev; __syncthreads();
  if (n == 0) { float s = 0.f; for (int t = 0; t < 256; ++t) s += red[t]; ssum = s; }
  __syncthreads();
  att[b * 256 + n] = ev / ssum;
}

__global__ void k_wm(const float* att, const float* mn, float* wm) {
  int b = blockIdx.x, d = threadIdx.x;  // block 512
  float acc = 0.f;
  for (int n = 0; n < 256; ++n)
    acc += att[b * 256 + n] * mn[(size_t)(b * 256 + n) * 512 + d];
  wm[(size_t)b * 512 + d] = acc;
}

__global__ void k_hp_attn(const float* qh, const float* kh, const float* tape,
                          const float* beta_p, float* attn) {
  int b = blockIdx.x, t = threadIdx.x;  // block 128
  float s = 0.f;
  for (int r = 0; r < 64; ++r) s += qh[b * 64 + r] * kh[(size_t)(b * 128 + t) * 64 + r];
  s *= log1pf(expf(beta_p[0]));         // softplus(beta)
  float asum = 0.f;
  for (int r = 0; r < 64; ++r) asum += fabsf(tape[(size_t)(b * 128 + t) * 64 + r]);
  if (asum < 1e-6f) s = -1e9f;
  __shared__ float red[128];
  __shared__ float smax, ssum;
  red[t] = s; __syncthreads();
  if (t == 0) { float m = red[0]; for (int u = 1; u < 128; ++u) m = fmaxf(m, red[u]); smax = m; }
  __syncthreads();
  float ev = expf(s - smax);
  red[t] = ev; __syncthreads();
  if (t == 0) { float su = 0.f; for (int u = 0; u < 128; ++u) su += red[u]; ssum = su; }
  __syncthreads();
  float a = ev / ssum;
  if (!(a == a)) a = 0.f;               // nan_to_num
  attn[b * 128 + t] = a;
}

__global__ void k_retr(const float* attn, const float* tapev, float* retr) {
  int b = blockIdx.x, r = threadIdx.x;  // block 64
  float acc = 0.f;
  for (int t = 0; t < 128; ++t)
    acc += attn[b * 128 + t] * tapev[(size_t)(b * 128 + t) * 64 + r];
  retr[b * 64 + r] = acc;
}

__global__ void k_hp_gate(const float* wm, const float* retrD, const float* W,
                          const float* bias, float* gx) {
  int b = blockIdx.x, d = threadIdx.x;  // block 512
  float acc = bias[d];
  for (int k = 0; k < 512; ++k) acc += wm[(size_t)b * 512 + k] * W[(size_t)k * 512 + d];
  for (int k = 0; k < 512; ++k) acc += retrD[(size_t)b * 512 + k] * W[(size_t)(512 + k) * 512 + d];
  float gate = sigf(acc);
  gx[(size_t)b * 512 + d] = wm[(size_t)b * 512 + d] + gate * retrD[(size_t)b * 512 + d];
}

// ---------------------------------------------------------------------------
extern "C" void kernel_launch(void* const* d_in, const int* in_sizes, int n_in,
                              void* d_out, int out_size, void* d_ws, size_t ws_size,
                              hipStream_t stream) {
  (void)in_sizes; (void)n_in; (void)out_size; (void)ws_size;
  const int*   x_t      = (const int*)d_in[0];
  const int*   role_id  = (const int*)d_in[1];
  const float* m_prev   = (const float*)d_in[2];
  const float* M_prev   = (const float*)d_in[3];
  const float* tape     = (const float*)d_in[4];
  const float* emb      = (const float*)d_in[5];
  const float* role_emb = (const float*)d_in[6];
  const float* neuron_emb = (const float*)d_in[7];
  const float* norm_g   = (const float*)d_in[8];
  const float* norm_b   = (const float*)d_in[9];
  const float* router_w = (const float*)d_in[10];
  const float* gm_proj_w = (const float*)d_in[11];
  const float* gm_proj_b = (const float*)d_in[12];
  const float* gm_gate_w = (const float*)d_in[13];
  const float* gm_gate_b = (const float*)d_in[14];
  const float* intra_w  = (const float*)d_in[15];
  const float* intra_b  = (const float*)d_in[16];
  const float* hq_w = (const float*)d_in[17];
  const float* hq_b = (const float*)d_in[18];
  const float* hk_w = (const float*)d_in[19];
  const float* hk_b = (const float*)d_in[20];
  const float* hv_w = (const float*)d_in[21];
  const float* hv_b = (const float*)d_in[22];
  const float* hv_ln_g = (const float*)d_in[23];
  const float* hv_ln_b = (const float*)d_in[24];
  const float* h_out_w = (const float*)d_in[25];
  const float* h_out_b = (const float*)d_in[26];
  const float* decay_w = (const float*)d_in[27];
  const float* decay_b = (const float*)d_in[28];
  const float* ops_w1 = (const float*)d_in[29];
  const float* ops_b1 = (const float*)d_in[30];
  const float* ops_w2 = (const float*)d_in[31];
  const float* ops_b2 = (const float*)d_in[32];
  const float* op_logits = (const float*)d_in[33];
  const float* f_w = (const float*)d_in[34];
  const float* f_b = (const float*)d_in[35];
  const float* i_w = (const float*)d_in[36];
  const float* i_b = (const float*)d_in[37];
  const float* mlp_w1 = (const float*)d_in[38];
  const float* mlp_b1 = (const float*)d_in[39];
  const float* mlp_w2 = (const float*)d_in[40];
  const float* mlp_b2 = (const float*)d_in[41];
  const float* ro_norm_g = (const float*)d_in[42];
  const float* ro_norm_b = (const float*)d_in[43];
  const float* readout_q = (const float*)d_in[44];
  const float* hp_compress = (const float*)d_in[45];
  const float* hp_decompress = (const float*)d_in[46];
  const float* hp_q = (const float*)d_in[47];
  const float* hp_k = (const float*)d_in[48];
  const float* hp_v = (const float*)d_in[49];
  const float* hp_gate_w = (const float*)d_in[50];
  const float* hp_gate_b = (const float*)d_in[51];
  const float* hp_beta = (const float*)d_in[52];
  const float* motor_g = (const float*)d_in[53];
  const float* motor_b = (const float*)d_in[54];

  float* ws = (float*)d_ws;
  float* out_ro = (float*)d_out;          // [32,512]
  float* out_m  = out_ro + 16384;         // [32,256,512]
  float* out_M  = out_m + 4194304;        // [32,16,64,64]

  // workspace layout (floats)
  float* token   = ws + 0;                // 16384
  float* rwbuf   = ws + 16384;            // 512
  float* activ   = ws + 16896;            // 512
  float* mnorm   = ws + 17408;            // 4194304 (reused for m_n later)
  float* grp     = ws + 4211712;          // 262144
  float* glob    = ws + 4473856;          // 16384
  float* routed  = ws + 4490240;          // 16384
  float* inter   = ws + 4506624;          // 262144
  float* intra   = ws + 4768768;          // 4194304
  float* qbuf    = ws + 8963072;          // 524288
  float* kbuf    = ws + 9487360;          // 524288
  float* vbuf    = ws + 10011648;         // 524288
  float* hebb    = ws + 10535936;         // 524288
  float* hebbout = ws + 11060224;         // 4194304
  float* comb    = ws + 15254528;         // 20971520 (reused as h2)
  float* h1      = ws + 36226048;         // 20971520
  float* mlph    = ws + 57197568;         // 8388608 (reused: f at +0, i at +4194304)
  float* mlpout  = ws + 65586176;         // 4194304
  float* att     = ws + 69780480;         // 8192
  float* wmb     = ws + 69788672;         // 16384
  float* t1      = ws + 69805056;         // 2048
  float* qh      = ws + 69807104;         // 2048
  float* kh      = ws + 69809152;         // 262144
  float* tapev   = ws + 70071296;         // 262144
  float* sattn   = ws + 70333440;         // 4096
  float* retr    = ws + 70337536;         // 2048
  float* retrD   = ws + 70339584;         // 16384
  float* gx      = ws + 70355968;         // 16384
  float* fbuf    = mlph;                  // alias after mlp2
  float* ibuf    = mlph + 4194304;        // alias
  float* h2      = comb;                  // alias after ops_w1
  float* m_n     = mnorm;                 // alias after ops_update

  // 1. token + router softmax + top-2
  k_token_router<<<32, 512, 0, stream>>>(x_t, role_id, emb, role_emb, router_w,
                                         token, rwbuf, activ);
  // 2. m_norm = LN(m_prev * mask)
  ln_k<<<8192, 256, 0, stream>>>(m_prev, activ, norm_g, norm_b, mnorm, 512, 1e-5f);
  // 3-5. workspace message pass
  k_grp<<<512, 256, 0, stream>>>(mnorm, grp);
  k_glob<<<32, 256, 0, stream>>>(grp, rwbuf, glob);
  naive_mm<<<32, 512, 0, stream>>>(glob, gm_proj_w, gm_proj_b, routed, 512, 512, 0);
  k_inter_gate<<<512, 512, 0, stream>>>(grp, routed, gm_gate_w, gm_gate_b, inter);
  // 6. intra: per-group [32x8192] @ [8192x8192]  (WMMA)
  gemm_wmma<<<dim3(128, 1, 16), 256, 0, stream>>>(
      mnorm, intra_w, intra_b, intra, 32, 8192, 8192,
      131072, 8192, 131072, 8192LL, 8192LL, 0);
  // 7. q/k/v projections (WMMA) + normalizations
  gemm_wmma<<<dim3(1, 256, 1), 256, 0, stream>>>(mnorm, hq_w, hq_b, qbuf,
      8192, 64, 512, 512, 64, 64, 0LL, 0LL, 0);
  rownorm_k<<<8192, 64, 0, stream>>>(qbuf, 1e-5f);
  gemm_wmma<<<dim3(1, 256, 1), 256, 0, stream>>>(mnorm, hk_w, hk_b, kbuf,
      8192, 64, 512, 512, 64, 64, 0LL, 0LL, 0);
  rownorm_k<<<8192, 64, 0, stream>>>(kbuf, 1e-5f);
  gemm_wmma<<<dim3(1, 256, 1), 256, 0, stream>>>(mnorm, hv_w, hv_b, vbuf,
      8192, 64, 512, 512, 64, 64, 0LL, 0LL, 0);
  ln_k<<<8192, 256, 0, stream>>>(vbuf, nullptr, hv_ln_g, hv_ln_b, vbuf, 64, 1e-5f);
  // 8. Hebbian fast-weight update; writes M_next
  k_hebbian<<<512, 256, 0, stream>>>(mnorm, qbuf, kbuf, vbuf, M_prev,
                                     decay_w, decay_b, activ, out_M, hebb);
  // 9. hebb_out (WMMA)
  gemm_wmma<<<dim3(8, 256, 1), 256, 0, stream>>>(hebb, h_out_w, h_out_b, hebbout,
      8192, 512, 64, 64, 512, 512, 0LL, 0LL, 0);
  // 10. combined
  k_concat<<<8192, 512, 0, stream>>>(token, neuron_emb, inter, intra,
                                     hebbout, mnorm, comb);
  // 11. MLP (fused gelu), then f/i gates (fused sigmoid), ops chain (WMMA)
  gemm_wmma<<<dim3(16, 256, 1), 256, 0, stream>>>(comb, mlp_w1, mlp_b1, mlph,
      8192, 1024, 2560, 2560, 1024, 1024, 0LL, 0LL, 2);
  gemm_wmma<<<dim3(8, 256, 1), 256, 0, stream>>>(mlph, mlp_w2, mlp_b2, mlpout,
      8192, 512, 1024, 1024, 512, 512, 0LL, 0LL, 0);
  gemm_wmma<<<dim3(8, 256, 1), 256, 0, stream>>>(comb, f_w, f_b, fbuf,
      8192, 512, 2560, 2560, 512, 512, 0LL, 0LL, 1);
  gemm_wmma<<<dim3(8, 256, 1), 256, 0, stream>>>(comb, i_w, i_b, ibuf,
      8192, 512, 2560, 2560, 512, 512, 0LL, 0LL, 1);
  gemm_wmma<<<dim3(40, 256, 1), 256, 0, stream>>>(comb, ops_w1, ops_b1, h1,
      8192, 2560, 2560, 2560, 2560, 2560, 0LL, 0LL, 2);
  gemm_wmma<<<dim3(40, 256, 1), 256, 0, stream>>>(h1, ops_w2, ops_b2, h2,
      8192, 2560, 2560, 2560, 2560, 2560, 0LL, 0LL, 0);
  // 12. state update -> m_next
  k_ops_update<<<8192, 512, 0, stream>>>(h2, mlpout, fbuf, ibuf, mnorm, m_prev,
                                         activ, op_logits, out_m);
  // 13. readout LN + attention pooling
  ln_k<<<8192, 256, 0, stream>>>(out_m, nullptr, ro_norm_g, ro_norm_b, m_n, 512, 1e-5f);
  k_readout_att<<<32, 256, 0, stream>>>(m_n, readout_q, att);
  k_wm<<<32, 512, 0, stream>>>(att, m_n, wmb);
  // 14. hippocampal retrieval
  naive_mm<<<32, 64, 0, stream>>>(wmb, hp_compress, nullptr, t1, 512, 64, 0);
  naive_mm<<<32, 64, 0, stream>>>(t1, hp_q, nullptr, qh, 64, 64, 0);
  rownorm_k<<<32, 64, 0, stream>>>(qh, 1e-12f);
  naive_mm<<<4096, 64, 0, stream>>>(tape, hp_k, nullptr, kh, 64, 64, 0);
  rownorm_k<<<4096, 64, 0, stream>>>(kh, 1e-12f);
  naive_mm<<<4096, 64, 0, stream>>>(tape, hp_v, nullptr, tapev, 64, 64, 0);
  k_hp_attn<<<32, 128, 0, stream>>>(qh, kh, tape, hp_beta, sattn);
  k_retr<<<32, 64, 0, stream>>>(sattn, tapev, retr);
  naive_mm<<<32, 512, 0, stream>>>(retr, hp_decompress, nullptr, retrD, 64, 512, 0);
  k_hp_gate<<<32, 512, 0, stream>>>(wmb, retrD, hp_gate_w, hp_gate_b, gx);
  ln_k<<<32, 256, 0, stream>>>(gx, nullptr, motor_g, motor_b, out_ro, 512, 1e-5f);
}